// GraphAttentionLayer_76124000354340
// MI455X (gfx1250) — compile-verified
//
#include <hip/hip_runtime.h>
#include <hip/hip_bf16.h>

#define B_   2
#define N_   8192
#define F_   64
#define E_   262144
#define ALPHA 0.2f

typedef float v2f __attribute__((ext_vector_type(2)));
typedef float v8f __attribute__((ext_vector_type(8)));

// ---- ordered-uint encoding so we can do float atomic max with atomicMax(u32) ----
__device__ __forceinline__ unsigned ford(float f) {
    unsigned u = __float_as_uint(f);
    return (u & 0x80000000u) ? ~u : (u | 0x80000000u);
}
__device__ __forceinline__ float funord(unsigned u) {
    return (u & 0x80000000u) ? __uint_as_float(u & 0x7FFFFFFFu) : __uint_as_float(~u);
}

// ---------------- 0: zero init (rowmax/rowsum + output accumulator) ----------------
__global__ void k_zero(unsigned* __restrict__ rmax, float* __restrict__ rsum,
                       float* __restrict__ out) {
    int t = blockIdx.x * blockDim.x + threadIdx.x;   // grid covers B*N*F
    if (t < B_ * N_) { rmax[t] = 0u; rsum[t] = 0.0f; }
    out[t] = 0.0f;
}

// ---------------- 1: Wh = h @ W^T via V_WMMA_F32_16X16X4_F32 ----------------
// One wave computes one 16x16 tile of Wh. K=64 -> 16 chained K=4 WMMAs.
// A (16x4 f32): lane<16 holds M=lane, {K,K+1}; lane>=16 holds M=lane-16, {K+2,K+3}.
// B (4x16 f32): mirrored with N across lanes. B[k][n] = W[n][k] (W is row-major (out,in)).
__global__ void k_gemm_wmma(const float* __restrict__ h, const float* __restrict__ W,
                            float* __restrict__ Wh) {
    int wave = blockIdx.x * (blockDim.x >> 5) + (threadIdx.x >> 5);
    int lane = threadIdx.x & 31;
    // waves per batch: (N/16) * (F/16) = 512*4
    int b  = wave / (512 * 4);
    int t  = wave % (512 * 4);
    int tm = t >> 2;          // M tile 0..511
    int tn = t & 3;           // N tile 0..3
    const float* hb  = h  + (size_t)b * N_ * F_;
    float*       whb = Wh + (size_t)b * N_ * F_;
    int m0 = tm * 16, n0 = tn * 16;
    int lr = lane & 15;
    int hi = lane >> 4;

    v8f c = {0.f, 0.f, 0.f, 0.f, 0.f, 0.f, 0.f, 0.f};
    #pragma unroll
    for (int k = 0; k < F_; k += 4) {
        v2f a  = *(const v2f*)(hb + (size_t)(m0 + lr) * F_ + k + 2 * hi);
        v2f bb = *(const v2f*)(W  + (size_t)(n0 + lr) * F_ + k + 2 * hi);
        c = __builtin_amdgcn_wmma_f32_16x16x4_f32(false, a, false, bb,
                                                  (short)0, c, false, false);
    }
    // C/D layout: VGPR r, lanes<16 -> M=r, lanes>=16 -> M=8+r; N = lane%16.
    #pragma unroll
    for (int r = 0; r < 8; ++r) {
        int row = m0 + r + hi * 8;
        whb[(size_t)row * F_ + n0 + lr] = c[r];
    }
}

// ---------------- 2: s1[n] = Wh[n].a[:64], s2[n] = Wh[n].a[64:] ----------------
__global__ void k_proj(const float* __restrict__ Wh, const float* __restrict__ a,
                       float* __restrict__ s1, float* __restrict__ s2) {
    int n = blockIdx.x * blockDim.x + threadIdx.x;   // over B*N
    if (n >= B_ * N_) return;
    const float* row = Wh + (size_t)n * F_;
    float acc1 = 0.f, acc2 = 0.f;
    #pragma unroll
    for (int o = 0; o < F_; ++o) {
        float v = row[o];
        acc1 += v * a[o];
        acc2 += v * a[F_ + o];
    }
    s1[n] = acc1;
    s2[n] = acc2;
}

// ---------------- 3: per-edge logits + row max ----------------
__global__ void k_edge_max(const long long* __restrict__ ei, const float* __restrict__ ew,
                           const float* __restrict__ s1, const float* __restrict__ s2,
                           float* __restrict__ ebuf, unsigned* __restrict__ rmax) {
    int t = blockIdx.x * blockDim.x + threadIdx.x;   // over B*E
    if (t >= B_ * E_) return;
    int b = t / E_, k = t % E_;
    int src = (int)ei[k];
    int dst = (int)ei[E_ + k];
    float e = s1[b * N_ + src] + s2[b * N_ + dst];
    e = (e > 0.f) ? e : ALPHA * e;       // leaky relu
    e *= ew[k];
    ebuf[t] = e;
    if (e != 0.0f)                        // zeros are masked in the reference
        atomicMax(&rmax[b * N_ + src], ford(e));
}

// ---------------- 4: p = exp(e - max), row sums ----------------
__global__ void k_edge_exp(const long long* __restrict__ ei, const unsigned* __restrict__ rmax,
                           float* __restrict__ ebuf, float* __restrict__ rsum) {
    int t = blockIdx.x * blockDim.x + threadIdx.x;   // over B*E
    if (t >= B_ * E_) return;
    int b = t / E_, k = t % E_;
    float e = ebuf[t];
    float p = 0.0f;
    if (e != 0.0f) {
        int src = (int)ei[k];
        float m = funord(rmax[b * N_ + src]);
        p = expf(e - m);
        atomicAdd(&rsum[b * N_ + src], p);
    }
    ebuf[t] = p;
}

// ---------------- 5: scatter coef * Wh[dst] into out[src] (wave per edge) ----------------
__global__ void k_edge_scatter(const long long* __restrict__ ei, const float* __restrict__ ebuf,
                               const float* __restrict__ rsum, const float* __restrict__ Wh,
                               float* __restrict__ out) {
    int wave = blockIdx.x * (blockDim.x >> 5) + (threadIdx.x >> 5);
    int lane = threadIdx.x & 31;
    if (wave >= B_ * E_) return;
    float p = ebuf[wave];
    if (p == 0.0f) return;
    int b = wave / E_, k = wave % E_;
    int src = (int)ei[k];
    int dst = (int)ei[E_ + k];
    float coef = p / rsum[b * N_ + src];   // rsum > 0 since p > 0 contributed
    const float* whd = Wh  + ((size_t)b * N_ + dst) * F_;
    float*       ac  = out + ((size_t)b * N_ + src) * F_;
    atomicAdd(&ac[lane],      coef * whd[lane]);
    atomicAdd(&ac[lane + 32], coef * whd[lane + 32]);
}

// ---------------- 6: in-place ELU ----------------
__global__ void k_elu(float* __restrict__ out) {
    int t = blockIdx.x * blockDim.x + threadIdx.x;   // over B*N*F
    float x = out[t];
    out[t] = (x > 0.f) ? x : (expf(x) - 1.0f);
}

extern "C" void kernel_launch(void* const* d_in, const int* in_sizes, int n_in,
                              void* d_out, int out_size, void* d_ws, size_t ws_size,
                              hipStream_t stream) {
    const float*     h  = (const float*)d_in[0];       // (B,N,64)
    const float*     W  = (const float*)d_in[1];       // (64,64)
    const float*     a  = (const float*)d_in[2];       // (128,)
    const long long* ei = (const long long*)d_in[3];   // (2,E) int64
    const float*     ew = (const float*)d_in[4];       // (E,)
    float*           out = (float*)d_out;              // (B,N,64)

    // workspace carve-up
    float*    Wh   = (float*)d_ws;                     // B*N*F
    float*    s1   = Wh + (size_t)B_ * N_ * F_;        // B*N
    float*    s2   = s1 + (size_t)B_ * N_;             // B*N
    unsigned* rmax = (unsigned*)(s2 + (size_t)B_ * N_);// B*N
    float*    rsum = (float*)(rmax + (size_t)B_ * N_); // B*N
    float*    ebuf = rsum + (size_t)B_ * N_;           // B*E

    // 0: init (covers B*N*F elements exactly; B*N subset inits rmax/rsum)
    k_zero<<<(B_ * N_ * F_) / 256, 256, 0, stream>>>(rmax, rsum, out);

    // 1: WMMA GEMM — B * 512 * 4 = 4096 waves, 4 waves/block of 128 threads
    k_gemm_wmma<<<(B_ * 512 * 4) / 4, 128, 0, stream>>>(h, W, Wh);

    // 2: attention projections
    k_proj<<<(B_ * N_ + 255) / 256, 256, 0, stream>>>(Wh, a, s1, s2);

    // 3: logits + row max
    k_edge_max<<<(B_ * E_ + 255) / 256, 256, 0, stream>>>(ei, ew, s1, s2, ebuf, rmax);

    // 4: exp + row sums
    k_edge_exp<<<(B_ * E_ + 255) / 256, 256, 0, stream>>>(ei, rmax, ebuf, rsum);

    // 5: scatter — wave per edge, 8 waves per 256-thread block
    k_edge_scatter<<<(B_ * E_) / 8, 256, 0, stream>>>(ei, ebuf, rsum, Wh, out);

    // 6: ELU in place
    k_elu<<<(B_ * N_ * F_) / 256, 256, 0, stream>>>(out);
}